// ST_Affine_54468775248451
// MI455X (gfx1250) — compile-verified
//
#include <hip/hip_runtime.h>
#include <cstddef>
#include <cstdint>

// STN: affine grid + bilinear grid_sample (padding=zeros, align_corners=False)
// x: [16,64,256,256] fp32, theta: [16,6] fp32 -> out: [16,64,256,256] fp32
//
// Each block owns a 32x8 output tile of one image. The affine warp maps the
// tile to a small source bounding box (exact: affine => extrema at tile
// corners). Per round, the bbox of 4 channels is staged into 4 LDS slots with
// GLOBAL_LOAD_ASYNC_TO_LDS_B128 (ASYNCcnt), then each thread does 4 LDS reads
// + 3 FMAs + 1 non-temporal store per channel. Index math is computed once per
// pixel and amortized over all 64 channels. Uniform fallback to direct global
// gathers if the bbox exceeds the LDS slot.

#define STN_B 16
#define STN_C 64
#define STN_H 256
#define STN_W 256
#define STN_HW (STN_H * STN_W)

#define TILE_W 32
#define TILE_H 8
#define NTHREADS (TILE_W * TILE_H)   // 256 threads = 8 wave32
#define CHSLOT 2048                  // floats per channel slot
#define CBATCH 4                     // channels staged per sync round
#define LDS_FLOATS (CHSLOT * CBATCH) // 8192 floats = 32 KB static LDS

#if __has_builtin(__builtin_amdgcn_global_load_async_to_lds_b32)
#define HAVE_ASYNC_LDS 1
#else
#define HAVE_ASYNC_LDS 0
#endif
#if __has_builtin(__builtin_amdgcn_global_load_async_to_lds_b128)
#define HAVE_ASYNC_B128 1
#else
#define HAVE_ASYNC_B128 0
#endif

typedef int stn_v4i __attribute__((__vector_size__(4 * sizeof(int))));
typedef __attribute__((address_space(1))) int*      as1_i32p;
typedef __attribute__((address_space(3))) int*      as3_i32p;
typedef __attribute__((address_space(1))) stn_v4i*  as1_v4ip;
typedef __attribute__((address_space(3))) stn_v4i*  as3_v4ip;

__device__ __forceinline__ void stn_src_coord(const float t[6], float wf, float hf,
                                              float& ix, float& iy) {
  // Single shared formula: tile-corner bbox math and per-thread coords agree
  // bitwise (floor/clamp are monotone), so bbox containment is exact.
  const float gx = (wf + 0.5f) * (2.0f / STN_W) - 1.0f;
  const float gy = (hf + 0.5f) * (2.0f / STN_H) - 1.0f;
  const float sx = fmaf(t[0], gx, fmaf(t[1], gy, t[2]));
  const float sy = fmaf(t[3], gx, fmaf(t[4], gy, t[5]));
  ix = fmaf(sx, 0.5f * STN_W, 0.5f * STN_W - 0.5f);  // ((sx+1)*W - 1) * 0.5
  iy = fmaf(sy, 0.5f * STN_H, 0.5f * STN_H - 0.5f);
}

__global__ __launch_bounds__(NTHREADS) void stn_affine_sample_kernel(
    const float* __restrict__ x, const float* __restrict__ theta,
    float* __restrict__ out) {
  __shared__ float lbuf[LDS_FLOATS];

  const int b  = blockIdx.z;
  const int w0 = blockIdx.x * TILE_W;
  const int h0 = blockIdx.y * TILE_H;
  const int tw = threadIdx.x & (TILE_W - 1);   // lane -> column (coalesced)
  const int th = threadIdx.x >> 5;             // wave  -> row
  const int w  = w0 + tw;
  const int h  = h0 + th;

  float t[6];
#pragma unroll
  for (int i = 0; i < 6; ++i) t[i] = theta[b * 6 + i];  // uniform -> s_load

  // ---- per-pixel source coordinate, weights, clamped corner indices ----
  float ix, iy;
  stn_src_coord(t, (float)w, (float)h, ix, iy);
  const float fx0 = floorf(ix), fy0 = floorf(iy);
  const float tx = ix - fx0, ty = iy - fy0;
  const int x0 = (int)fx0, y0 = (int)fy0;
  const int x1 = x0 + 1,   y1 = y0 + 1;

  const float mx0 = ((unsigned)x0 < (unsigned)STN_W) ? 1.0f : 0.0f;
  const float mx1 = ((unsigned)x1 < (unsigned)STN_W) ? 1.0f : 0.0f;
  const float my0 = ((unsigned)y0 < (unsigned)STN_H) ? 1.0f : 0.0f;
  const float my1 = ((unsigned)y1 < (unsigned)STN_H) ? 1.0f : 0.0f;

  const float omtx = 1.0f - tx, omty = 1.0f - ty;
  const float w00 = omtx * omty * (mx0 * my0);
  const float w01 = tx   * omty * (mx1 * my0);
  const float w10 = omtx * ty   * (mx0 * my1);
  const float w11 = tx   * ty   * (mx1 * my1);

  const int cx0 = min(max(x0, 0), STN_W - 1);
  const int cx1 = min(max(x1, 0), STN_W - 1);
  const int cy0 = min(max(y0, 0), STN_H - 1);
  const int cy1 = min(max(y1, 0), STN_H - 1);

  // ---- block-uniform source bounding box from the 4 tile corners ----
  float ixa, iya, ixb, iyb, ixc, iyc, ixd, iyd;
  stn_src_coord(t, (float)w0,                (float)h0,                ixa, iya);
  stn_src_coord(t, (float)(w0 + TILE_W - 1), (float)h0,                ixb, iyb);
  stn_src_coord(t, (float)w0,                (float)(h0 + TILE_H - 1), ixc, iyc);
  stn_src_coord(t, (float)(w0 + TILE_W - 1), (float)(h0 + TILE_H - 1), ixd, iyd);
  const float minix = fminf(fminf(ixa, ixb), fminf(ixc, ixd));
  const float maxix = fmaxf(fmaxf(ixa, ixb), fmaxf(ixc, ixd));
  const float miniy = fminf(fminf(iya, iyb), fminf(iyc, iyd));
  const float maxiy = fmaxf(fmaxf(iya, iyb), fmaxf(iyc, iyd));

  int bx0 = min(max((int)floorf(minix), 0), STN_W - 1);
  const int bx1 = min(max((int)floorf(maxix) + 1, bx0), STN_W - 1);
  const int by0 = min(max((int)floorf(miniy), 0), STN_H - 1);
  const int by1 = min(max((int)floorf(maxiy) + 1, by0), STN_H - 1);
  bx0 &= ~3;                                   // 16B-align global row starts
  const int bw = (bx1 - bx0 + 1 + 3) & ~3;     // row length, multiple of 4
  // Note: 256 - bx0 is a multiple of 4, so bx0 + bw <= 256: the rounded row
  // never reads past the plane's right edge (no out-of-allocation access).
  const int bh = by1 - by0 + 1;
  const int area = bw * bh;                    // block-uniform

  const float* __restrict__ pin  = x   + (size_t)b * STN_C * STN_HW;
  float* __restrict__       pout = out + (size_t)b * STN_C * STN_HW + h * STN_W + w;

  if (area <= CHSLOT) {
    // LDS gather indices, clamped into the bbox (1-ulp safety: a clamped
    // neighbor can only be hit when its bilinear weight is ~0).
    const int ly0 = min(max(cy0 - by0, 0), bh - 1);
    const int ly1 = min(max(cy1 - by0, 0), bh - 1);
    const int lx0 = min(max(cx0 - bx0, 0), bw - 1);
    const int lx1 = min(max(cx1 - bx0, 0), bw - 1);
    const int l00 = ly0 * bw + lx0;
    const int l01 = ly0 * bw + lx1;
    const int l10 = ly1 * bw + lx0;
    const int l11 = ly1 * bw + lx1;

    for (int cb = 0; cb < STN_C; cb += CBATCH) {
      // stage CBATCH channel bboxes into CBATCH LDS slots (one sync round)
#pragma unroll
      for (int u = 0; u < CBATCH; ++u) {
        const float* plane = pin + (size_t)(cb + u) * STN_HW;
        float* lslot = &lbuf[u * CHSLOT];
        for (int r = th; r < bh; r += TILE_H) {
          const float* grow = plane + (by0 + r) * STN_W + bx0;
          float* lrow = lslot + r * bw;
#if HAVE_ASYNC_B128
          for (int col = tw * 4; col < bw; col += TILE_W * 4)   // 16B per lane
            __builtin_amdgcn_global_load_async_to_lds_b128(
                (as1_v4ip)(grow + col), (as3_v4ip)(lrow + col), 0, 0);
#elif HAVE_ASYNC_LDS
          for (int col = tw; col < bw; col += TILE_W)
            __builtin_amdgcn_global_load_async_to_lds_b32(
                (as1_i32p)(grow + col), (as3_i32p)(lrow + col), 0, 0);
#else
          for (int col = tw; col < bw; col += TILE_W)
            lrow[col] = grow[col];   // sync staging fallback (still LDS path)
#endif
        }
      }
#if HAVE_ASYNC_LDS || HAVE_ASYNC_B128
#if __has_builtin(__builtin_amdgcn_s_wait_asynccnt)
      __builtin_amdgcn_s_wait_asynccnt(0);
#else
      asm volatile("s_wait_asynccnt 0x0" ::: "memory");
#endif
#endif
      __syncthreads();  // all CBATCH slots visible to all waves

#pragma unroll
      for (int u = 0; u < CBATCH; ++u) {
        const float* lslot = &lbuf[u * CHSLOT];
        const float v = fmaf(w00, lslot[l00],
                        fmaf(w01, lslot[l01],
                        fmaf(w10, lslot[l10], w11 * lslot[l11])));
        __builtin_nontemporal_store(v, pout + (size_t)(cb + u) * STN_HW);
      }

      __syncthreads();  // safe to overwrite slots next round
    }
  } else {
    // pathological theta: proven direct-gather path (no barriers; uniform branch)
    const int o00 = cy0 * STN_W + cx0;
    const int o01 = cy0 * STN_W + cx1;
    const int o10 = cy1 * STN_W + cx0;
    const int o11 = cy1 * STN_W + cx1;
    for (int c = 0; c < STN_C; ++c) {
      const float* q = pin + (size_t)c * STN_HW;
      const float v = fmaf(w00, q[o00], fmaf(w01, q[o01], fmaf(w10, q[o10], w11 * q[o11])));
      __builtin_nontemporal_store(v, pout + (size_t)c * STN_HW);
    }
  }
}

extern "C" void kernel_launch(void* const* d_in, const int* in_sizes, int n_in,
                              void* d_out, int out_size, void* d_ws, size_t ws_size,
                              hipStream_t stream) {
  const float* x     = (const float*)d_in[0];   // [16,64,256,256]
  const float* theta = (const float*)d_in[1];   // [16,6]
  float* out         = (float*)d_out;           // [16,64,256,256]

  dim3 grid(STN_W / TILE_W, STN_H / TILE_H, STN_B);  // 8 x 32 x 16 = 4096 blocks
  dim3 block(NTHREADS);
  stn_affine_sample_kernel<<<grid, block, 0, stream>>>(x, theta, out);
}